// Masked_Model_51264729645284
// MI455X (gfx1250) — compile-verified
//
#include <hip/hip_runtime.h>

// Problem constants (reference: B=2048, IMG=64, C=3)
#define NPE   12288          // 64*64*3 elements per row
#define TPB   512            // threads per block (16 waves of 32)
#define C4N   (NPE / 4)      // 3072 float4 chunks per row
#define C4PT  (C4N / TPB)    // 6 float4 per thread
#define EPT   (NPE / TPB)    // 24 elements per thread
#define NWAVE (TPB / 32)     // 16 waves

#if __has_builtin(__builtin_amdgcn_global_load_async_to_lds_b128)
#define HAVE_ASYNC_LDS 1
#else
#define HAVE_ASYNC_LDS 0
#endif

// Pointer types matching the builtin's expected parameters:
// b128 transfer -> pointer to 4 x i32, global (AS1) source, LDS (AS3) dest.
typedef int v4i_cd5 __attribute__((ext_vector_type(4)));
typedef __attribute__((address_space(1))) v4i_cd5* gv4p_cd5;
typedef __attribute__((address_space(3))) v4i_cd5* lv4p_cd5;

__device__ __forceinline__ void wait_async0() {
#if __has_builtin(__builtin_amdgcn_s_wait_asynccnt)
  __builtin_amdgcn_s_wait_asynccnt(0);
#else
  asm volatile("s_wait_asynccnt 0" ::: "memory");
#endif
}

__global__ void __launch_bounds__(TPB)
mask_topk_kernel(const float* __restrict__ data,
                 const float* __restrict__ sal,
                 const int*   __restrict__ perm,
                 float*       __restrict__ out)
{
  __shared__ __align__(16) float ldsRow[NPE];            // 48 KB: saliency row
  __shared__ unsigned int        histW[NWAVE * 256];     // 16 KB: per-wave fixed-point bins
  __shared__ unsigned long long  scanA[256];             // 2 KB : merged suffix sums
  __shared__ float               redBuf[NWAVE];
  __shared__ float               sTotal;
  __shared__ int                 sSel;
  __shared__ unsigned long long  sGp;
  __shared__ unsigned int        sPrefix;
  __shared__ int                 sTieCnt;
  __shared__ int                 tieIdx[64];

  const int tid = threadIdx.x;
  const int b   = blockIdx.x;

  const float* salRow = sal  + (size_t)b * NPE;
  const int    pb     = perm[b];
  const float* dRow   = data + (size_t)b  * NPE;
  const float* rRow   = data + (size_t)pb * NPE;
  float*       oRow   = out  + (size_t)b  * NPE;

  // Warm L2 with the two data rows needed by the output phase (global_prefetch_b8),
  // overlapping with the whole selection computation.
#pragma unroll
  for (int k = 0; k < C4PT; ++k) {
    const int e0 = (tid + (k << 9)) << 2;
    __builtin_prefetch(dRow + e0, 0, 1);
    __builtin_prefetch(rRow + e0, 0, 1);
  }

  // ---------- Pass 1: saliency row -> LDS, and deterministic row total ----------
  float psum = 0.f;
#if HAVE_ASYNC_LDS
#pragma unroll
  for (int k = 0; k < C4PT; ++k) {
    const int c4 = tid + (k << 9);
    __builtin_amdgcn_global_load_async_to_lds_b128(
        (gv4p_cd5)(salRow + (c4 << 2)),
        (lv4p_cd5)(&ldsRow[c4 << 2]),
        0, 0);
  }
  wait_async0();
  __syncthreads();
#pragma unroll
  for (int k = 0; k < C4PT; ++k) {
    float4 v = reinterpret_cast<const float4*>(ldsRow)[tid + (k << 9)];
    psum += (v.x + v.y) + (v.z + v.w);
  }
#else
#pragma unroll
  for (int k = 0; k < C4PT; ++k) {
    const int c4 = tid + (k << 9);
    float4 v = reinterpret_cast<const float4*>(salRow)[c4];
    reinterpret_cast<float4*>(ldsRow)[c4] = v;
    psum += (v.x + v.y) + (v.z + v.w);
  }
#endif
  // fixed-pattern wave32 reduce, then fixed-order cross-wave combine -> deterministic
  for (int o = 16; o > 0; o >>= 1) psum += __shfl_down(psum, o, 32);
  if ((tid & 31) == 0) redBuf[tid >> 5] = psum;
  __syncthreads();
  if (tid == 0) {
    float t = 0.f;
    for (int w = 0; w < NWAVE; ++w) t += redBuf[w];
    sTotal = t;
  }
  __syncthreads();

  const float totalF = sTotal;
  // Fixed-point domain: q(v) = (u32)(v * 2^31/total). Integer atomics are
  // order-independent -> deterministic; suffix sums exact & monotone.
  const float scaleF = (totalF > 0.f) ? (0x1p31f / totalF) : 0.f;
  const unsigned long long QT =
      (unsigned long long)((0.3f * totalF) * scaleF);   // thresh in q-units

  // ---------- Radix-select: 4 rounds of 8-bit digits, MSB -> LSB ----------
  unsigned int       prefix = 0u;   // matched high bits (boundary key prefix)
  unsigned long long Gp     = 0ull; // q-mass strictly above prefix region

  for (int r = 0; r < 4; ++r) {
    for (int i = tid; i < NWAVE * 256; i += TPB) histW[i] = 0u;
    __syncthreads();

    unsigned int* myHist = &histW[(tid >> 5) << 8];   // wave-private bins
    const int dgSh = 24 - 8 * r;
#pragma unroll
    for (int k = 0; k < EPT; ++k) {
      const unsigned int key = __float_as_uint(ldsRow[tid + (k << 9)]);
      const bool match = (r == 0) || ((key >> (32 - 8 * r)) == prefix);
      if (match) {
        const unsigned int q = (unsigned int)(__uint_as_float(key) * scaleF);
        atomicAdd(&myHist[(key >> dgSh) & 255u], q);
      }
    }
    __syncthreads();

    if (tid < 256) {
      unsigned long long s = 0ull;
      for (int w = 0; w < NWAVE; ++w) s += (unsigned long long)histW[(w << 8) + tid];
      scanA[tid] = s;
    }
    __syncthreads();

    // suffix-inclusive sums: scanA[t] = sum of bins t..255 (descending digit order)
    for (int st = 1; st < 256; st <<= 1) {
      unsigned long long v = 0ull;
      if (tid < 256) v = scanA[tid] + ((tid + st < 256) ? scanA[tid + st] : 0ull);
      __syncthreads();
      if (tid < 256) scanA[tid] = v;
      __syncthreads();
    }

    if (tid == 0) sSel = -1;
    __syncthreads();
    if (tid < 256) {
      const unsigned long long incl  = Gp + scanA[tid];
      const unsigned long long inclN = (tid < 255) ? (Gp + scanA[tid + 1]) : Gp;
      if (incl >= QT && inclN < QT) sSel = tid;   // unique (integer monotone)
    }
    __syncthreads();
    if (tid == 0) {
      int d = sSel;
      if (d < 0) d = 0;                            // cum never reaches thresh
      sGp     = Gp + ((d < 255) ? scanA[d + 1] : 0ull);
      sPrefix = (prefix << 8) | (unsigned int)d;
    }
    __syncthreads();
    Gp     = sGp;
    prefix = sPrefix;
  }

  // ---------- Exact tie handling at the boundary key ----------
  const unsigned int Kstar = prefix;
  const unsigned long long qstar =
      (unsigned long long)(unsigned int)(__uint_as_float(Kstar) * scaleF);

  if (tid == 0) sTieCnt = 0;
  __syncthreads();
#pragma unroll
  for (int k = 0; k < EPT; ++k) {
    const int e = tid + (k << 9);
    if (__float_as_uint(ldsRow[e]) == Kstar) {
      const int sl = atomicAdd(&sTieCnt, 1);
      if (sl < 64) tieIdx[sl] = e;
    }
  }
  __syncthreads();
  const int  tieN   = sTieCnt;
  const int  c      = (tieN < 64) ? tieN : 64;
  const bool bigTie = (tieN > 64);   // deterministic fallback (astronomically rare)

  // ---------- Output: out = mask ? data[perm[b]] : data[b] ----------
#pragma unroll
  for (int k = 0; k < C4PT; ++k) {
    const int c4 = tid + (k << 9);
    const int e0 = c4 << 2;
    const float4 dv = reinterpret_cast<const float4*>(dRow)[c4];
    const float4 rv = reinterpret_cast<const float4*>(rRow)[c4];
    const float4 kv = reinterpret_cast<const float4*>(ldsRow)[c4];
    const float dd[4] = {dv.x, dv.y, dv.z, dv.w};
    const float rr[4] = {rv.x, rv.y, rv.z, rv.w};
    const float kk[4] = {kv.x, kv.y, kv.z, kv.w};
    float ov[4];
#pragma unroll
    for (int i = 0; i < 4; ++i) {
      const unsigned int key = __float_as_uint(kk[i]);
      bool m;
      if (key > Kstar)      m = true;     // strictly above boundary: always masked
      else if (key < Kstar) m = false;    // strictly below: never masked
      else if (bigTie)      m = (Gp + qstar) < QT;
      else {
        int rk = 0;                        // rank among equal keys by original index
        for (int t = 0; t < c; ++t) rk += (tieIdx[t] < (e0 + i)) ? 1 : 0;
        m = (Gp + (unsigned long long)(rk + 1) * qstar) < QT;
      }
      ov[i] = m ? rr[i] : dd[i];
    }
    float4 o4;
    o4.x = ov[0]; o4.y = ov[1]; o4.z = ov[2]; o4.w = ov[3];
    reinterpret_cast<float4*>(oRow)[c4] = o4;
  }
}

extern "C" void kernel_launch(void* const* d_in, const int* in_sizes, int n_in,
                              void* d_out, int out_size, void* d_ws, size_t ws_size,
                              hipStream_t stream) {
  const float* data = (const float*)d_in[0];
  const float* sal  = (const float*)d_in[1];
  const int*   perm = (const int*)d_in[2];
  float*       out  = (float*)d_out;

  const int Brows = in_sizes[0] / NPE;   // 2048
  dim3 grid(Brows), block(TPB);
  hipLaunchKernelGGL(mask_topk_kernel, grid, block, 0, stream,
                     data, sal, perm, out);

  (void)n_in; (void)out_size; (void)d_ws; (void)ws_size;
}